// HeteroGNN_65386582114576
// MI455X (gfx1250) — compile-verified
//
#include <hip/hip_runtime.h>

// Problem constants (from reference)
constexpr int kNG = 100000;
constexpr int kND = 50000;
constexpr int kNC = 50000;
constexpr int kH  = 128;

typedef __attribute__((ext_vector_type(2))) float v2f;
typedef __attribute__((ext_vector_type(8))) float v8f;

// ---------------------------------------------------------------------------
// Scatter: per-relation segment sum + counts.
// One wave32 per edge; each lane moves 4 consecutive floats of the 128-f row.
// Tables (~102 MB total) fit in the 192 MB L2, so gathers/atomics stay on-chip.
// ---------------------------------------------------------------------------
__global__ __launch_bounds__(256)
void hgnn_scatter(const float* __restrict__ x, const int* __restrict__ src,
                  const int* __restrict__ dst, float* __restrict__ sum,
                  float* __restrict__ cnt, int nE) {
  const int lane = threadIdx.x & 31;
  const int wave = (blockIdx.x * blockDim.x + threadIdx.x) >> 5;
  const int nW   = (gridDim.x * blockDim.x) >> 5;
  for (int e = wave; e < nE; e += nW) {
    const int s = src[e];
    const int d = dst[e];
    const float4 v = *(const float4*)(x + (size_t)s * kH + lane * 4);
    float* p = sum + (size_t)d * kH + lane * 4;
    atomicAdd(p + 0, v.x);
    atomicAdd(p + 1, v.y);
    atomicAdd(p + 2, v.z);
    atomicAdd(p + 3, v.w);
    if (lane == 0) atomicAdd(cnt + d, 1.0f);
  }
}

// Wsum = A + B (128x128) so the gene GEMM needs only 3 terms.
__global__ void hgnn_addmat(const float* __restrict__ a, const float* __restrict__ b,
                            float* __restrict__ o, int n) {
  int i = blockIdx.x * blockDim.x + threadIdx.x;
  if (i < n) o[i] = a[i] + b[i];
}

// ---------------------------------------------------------------------------
// Fused multi-term GEMM:  Y = sum_t scale_t(A_t) @ W_t  (+ b0 + b1) (ReLU?)
// A_t: [rows,128] row-major; scale_t = 1/max(cnt_t,1) per row (segment mean).
// Block = 256 thr = 8 waves: 16 output rows; wave w -> 16x16 tile at col 16w.
// v_wmma_f32_16x16x4_f32, K loop of 32 steps, 16x128 A tile staged in LDS
// (stride 132 floats -> consecutive M-lanes hit distinct banks).
// In-place (Y aliasing A_t rows of this block) is safe: all reads precede
// the post-__syncthreads() store, and blocks only touch their own 16 rows.
// ---------------------------------------------------------------------------
__global__ __launch_bounds__(256)
void hgnn_gemm3(const float* __restrict__ A0, const float* __restrict__ c0,
                const float* __restrict__ W0,
                const float* __restrict__ A1, const float* __restrict__ c1,
                const float* __restrict__ W1,
                const float* __restrict__ A2, const float* __restrict__ c2,
                const float* __restrict__ W2,
                const float* __restrict__ b0, const float* __restrict__ b1,
                float* __restrict__ Y, int relu) {
  __shared__ float As[16 * 132];
  const int tid  = threadIdx.x;
  const int wv   = tid >> 5;
  const int lane = tid & 31;
  const int m    = lane & 15;            // M (A row) / N (B col) within tile
  const int koff = (lane >> 4) << 1;     // lanes 0-15: K+{0,1}; 16-31: K+{2,3}
  const int col  = (wv << 4) + m;        // output / B column
  const int row0 = blockIdx.x << 4;      // first output row of this block

  const float* Alist[3] = {A0, A1, A2};
  const float* Clist[3] = {c0, c1, c2};
  const float* Wlist[3] = {W0, W1, W2};

  v8f acc = {};
  for (int t = 0; t < 3; ++t) {
    const float* A = Alist[t];
    if (!A) continue;                    // uniform branch: EXEC stays all-1s
    const float* cp = Clist[t];
    const float* W  = Wlist[t];

    __syncthreads();
    for (int i = tid; i < 16 * kH; i += 256) {   // coalesced stage + mean scale
      const int r = i >> 7;
      const int c = i & (kH - 1);
      float v = A[(size_t)(row0 + r) * kH + c];
      if (cp) v *= 1.0f / fmaxf(cp[row0 + r], 1.0f);
      As[r * 132 + c] = v;
    }
    __syncthreads();

#pragma unroll 8
    for (int k0 = 0; k0 < kH; k0 += 4) {
      const int k = k0 + koff;
      v2f av, bv;
      av.x = As[m * 132 + k];                    // A frag: v0=K0/K2, v1=K1/K3
      av.y = As[m * 132 + k + 1];
      bv.x = W[(size_t)k * kH + col];            // B frag: same K split, N=lane
      bv.y = W[(size_t)(k + 1) * kH + col];
      // (neg_a, A, neg_b, B, c_mod, C, reuse_a, reuse_b)
      acc = __builtin_amdgcn_wmma_f32_16x16x4_f32(
          false, av, false, bv, (short)0, acc, false, false);
    }
  }

  const float bsum = (b0 ? b0[col] : 0.0f) + (b1 ? b1[col] : 0.0f);
  __syncthreads();  // all global reads of (possibly Y-aliased) A done
#pragma unroll
  for (int r = 0; r < 8; ++r) {
    float v = acc[r] + bsum;
    if (relu) v = fmaxf(v, 0.0f);
    // C/D layout: VGPR r -> M = r (lanes 0-15) / r+8 (lanes 16-31), N = lane&15
    Y[(size_t)(row0 + r + ((lane >> 4) << 3)) * kH + col] = v;
  }
}

// ---------------------------------------------------------------------------
extern "C" void kernel_launch(void* const* d_in, const int* in_sizes, int n_in,
                              void* d_out, int out_size, void* d_ws, size_t ws_size,
                              hipStream_t stream) {
  (void)n_in; (void)out_size; (void)ws_size;
  const float* EG = (const float*)d_in[0];   // emb_gene     [NG,128]
  const float* ED = (const float*)d_in[1];   // emb_disease  [ND,128]
  const float* EC = (const float*)d_in[2];   // emb_compound [NC,128]
  const float* Wl = (const float*)d_in[3];   // [2,4,128,128]
  const float* bl = (const float*)d_in[4];   // [2,4,128]
  const float* Wr = (const float*)d_in[5];   // [2,4,128,128]
  const int* eg_s  = (const int*)d_in[6];
  const int* eg_d  = (const int*)d_in[7];
  const int* ecd_s = (const int*)d_in[8];
  const int* ecd_d = (const int*)d_in[9];
  const int* ecg_s = (const int*)d_in[10];
  const int* ecg_d = (const int*)d_in[11];
  const int* egc_s = (const int*)d_in[12];
  const int* egc_d = (const int*)d_in[13];

  // Workspace layout (~155 MB): 4 mean-sum buffers + 4 count buffers + Wsum
  float* ws = (float*)d_ws;
  float* s0 = ws;                             // g<-g  (eg)   [NG,128]
  float* s1 = s0 + (size_t)kNG * kH;          // d<-c  (ecd)  [ND,128]
  float* s2 = s1 + (size_t)kND * kH;          // g<-c  (ecg)  [NG,128]
  float* s3 = s2 + (size_t)kNG * kH;          // c<-g  (egc)  [NC,128]
  float* c0 = s3 + (size_t)kNC * kH;
  float* c1 = c0 + kNG;
  float* c2 = c1 + kND;
  float* c3 = c2 + kNG;
  float* wsum = c3 + kNC;
  const size_t nAccum = (size_t)(2 * kNG + kND + kNC) * kH + (2 * kNG + kND + kNC);
  const size_t zero_bytes = nAccum * sizeof(float);

  float* OG = (float*)d_out;                  // output regions (concat order)
  float* OD = OG + (size_t)kNG * kH;
  float* OC = OD + (size_t)kND * kH;

  const float* xg = EG;
  const float* xd = ED;
  const float* xc = EC;

  for (int L = 0; L < 2; ++L) {
    hipMemsetAsync(ws, 0, zero_bytes, stream);
    hgnn_scatter<<<2048, 256, 0, stream>>>(xg, eg_s,  eg_d,  s0, c0, in_sizes[6]);
    hgnn_scatter<<<1024, 256, 0, stream>>>(xc, ecd_s, ecd_d, s1, c1, in_sizes[8]);
    hgnn_scatter<<<2048, 256, 0, stream>>>(xc, ecg_s, ecg_d, s2, c2, in_sizes[10]);
    hgnn_scatter<<<2048, 256, 0, stream>>>(xg, egc_s, egc_d, s3, c3, in_sizes[12]);

    const float* WlL = Wl + (size_t)L * 4 * kH * kH;
    const float* WrL = Wr + (size_t)L * 4 * kH * kH;
    const float* blL = bl + (size_t)L * 4 * kH;
    hgnn_addmat<<<(kH * kH + 255) / 256, 256, 0, stream>>>(WrL, WrL + 2 * kH * kH,
                                                           wsum, kH * kH);
    const int relu = (L == 0) ? 1 : 0;
    // genes: mean_gg@Wl0 + mean_cg@Wl2 + xg@(Wr0+Wr2) + b0 + b2
    hgnn_gemm3<<<kNG / 16, 256, 0, stream>>>(
        s0, c0, WlL,
        s2, c2, WlL + 2 * kH * kH,
        xg, nullptr, wsum,
        blL, blL + 2 * kH, OG, relu);
    // disease: mean_dc@Wl1 + xd@Wr1 + b1
    hgnn_gemm3<<<kND / 16, 256, 0, stream>>>(
        s1, c1, WlL + 1 * kH * kH,
        xd, nullptr, WrL + 1 * kH * kH,
        nullptr, nullptr, nullptr,
        blL + 1 * kH, nullptr, OD, relu);
    // compound: mean_cg@Wl3 + xc@Wr3 + b3
    hgnn_gemm3<<<kNC / 16, 256, 0, stream>>>(
        s3, c3, WlL + 3 * kH * kH,
        xc, nullptr, WrL + 3 * kH * kH,
        nullptr, nullptr, nullptr,
        blL + 3 * kH, nullptr, OC, relu);

    xg = OG; xd = OD; xc = OC;   // layer 2 reads/updates d_out in place
  }
}